// GeneralAttention_10230612099229
// MI455X (gfx1250) — compile-verified
//
#include <hip/hip_runtime.h>

// Problem constants (from reference): q[4,1024,64], k/v[4,4096,64]
#define B_      4
#define LQ      1024
#define LK      4096
#define DD      64
#define RUNS_   4
#define STEPS_  64
#define NCHAINS (B_ * LQ * RUNS_)          // 16384
#define NROWS   (B_ * LQ)                  // 4096
#define SCALE_  0.125f                     // 1/sqrt(64)

typedef float v2f __attribute__((ext_vector_type(2)));
typedef float v8f __attribute__((ext_vector_type(8)));

__device__ __forceinline__ unsigned rotl32(unsigned v, int n) {
  return (v << n) | (v >> (32 - n));
}

// JAX threefry2x32 (20 rounds), exact key schedule.
__device__ __forceinline__ void threefry2x32(unsigned k0, unsigned k1,
                                             unsigned c0, unsigned c1,
                                             unsigned& o0, unsigned& o1) {
  const unsigned ks2 = k0 ^ k1 ^ 0x1BD11BDAu;
  unsigned x0 = c0 + k0, x1 = c1 + k1;
  const int RA[4] = {13, 15, 26, 6};
  const int RB[4] = {17, 29, 16, 24};
#pragma unroll
  for (int r = 0; r < 4; r++) { x0 += x1; x1 = rotl32(x1, RA[r]); x1 ^= x0; }
  x0 += k1;  x1 += ks2 + 1u;
#pragma unroll
  for (int r = 0; r < 4; r++) { x0 += x1; x1 = rotl32(x1, RB[r]); x1 ^= x0; }
  x0 += ks2; x1 += k0 + 2u;
#pragma unroll
  for (int r = 0; r < 4; r++) { x0 += x1; x1 = rotl32(x1, RA[r]); x1 ^= x0; }
  x0 += k0;  x1 += k1 + 3u;
#pragma unroll
  for (int r = 0; r < 4; r++) { x0 += x1; x1 = rotl32(x1, RB[r]); x1 ^= x0; }
  x0 += k1;  x1 += ks2 + 4u;
#pragma unroll
  for (int r = 0; r < 4; r++) { x0 += x1; x1 = rotl32(x1, RA[r]); x1 ^= x0; }
  x0 += ks2; x1 += k0 + 5u;
  o0 = x0; o1 = x1;
}

// ---------------------------------------------------------------------------
// Phase 0: the scan's rng-split chain rng -> (rng', ka, kb) is identical for
// every chain (only the threefry *counter* differs per chain). Precompute the
// 64 per-step key pairs once (serial dependency, 1 thread) into 1KB of ws.
// ---------------------------------------------------------------------------
__global__ void compute_keys_kernel(unsigned* __restrict__ keys) {
  if (threadIdx.x == 0 && blockIdx.x == 0) {
    unsigned rng0 = 0u, rng1 = 42u;                 // jax.random.key(42)
    for (int t = 0; t < STEPS_; t++) {
      unsigned a0, a1, b0, b1, c0, c1;
      threefry2x32(rng0, rng1, 0u, 3u, a0, a1);
      threefry2x32(rng0, rng1, 1u, 4u, b0, b1);
      threefry2x32(rng0, rng1, 2u, 5u, c0, c1);
      keys[4 * t + 0] = c0;  // ka = (out0(2,5), out1(0,3))
      keys[4 * t + 1] = a1;
      keys[4 * t + 2] = b1;  // kb = (out1(1,4), out1(2,5))
      keys[4 * t + 3] = c1;
      rng0 = a0; rng1 = b0;  // rng' = (out0(0,3), out0(1,4))
    }
  }
}

// ---------------------------------------------------------------------------
// Phase 1: S[b,i,j] = SCALE * dot(q[b,i,:], k[b,j,:]) via V_WMMA_F32_16X16X4_F32
// One wave per 16x16 output tile; K=64 swept in 16 chunks of 4.
// ---------------------------------------------------------------------------
__global__ void scores_wmma_kernel(const float* __restrict__ q,
                                   const float* __restrict__ k,
                                   float* __restrict__ S) {
  const int wave = blockIdx.x * (blockDim.x >> 5) + (int)(threadIdx.x >> 5);
  const int lane = (int)(threadIdx.x & 31u);
  const int tilesPerBatch = (LQ / 16) * (LK / 16);   // 64 * 256
  const int b  = wave / tilesPerBatch;
  const int t  = wave % tilesPerBatch;
  const int it = t / (LK / 16);
  const int jt = t % (LK / 16);

  const float* qb = q + ((size_t)b * LQ + (size_t)it * 16) * DD;
  const float* kb = k + ((size_t)b * LK + (size_t)jt * 16) * DD;

  const int m  = lane & 15;            // row (A) / col (B) within tile
  const int ko = (lane >> 4) * 2;      // k sub-offset within chunk

  v8f acc = {};
#pragma unroll
  for (int kk = 0; kk < DD; kk += 4) {
    v2f a, bb;
    a.x  = qb[m * DD + kk + ko];
    a.y  = qb[m * DD + kk + ko + 1];
    bb.x = kb[m * DD + kk + ko];
    bb.y = kb[m * DD + kk + ko + 1];
    acc = __builtin_amdgcn_wmma_f32_16x16x4_f32(false, a, false, bb,
                                                (short)0, acc, false, false);
  }

  float* Sb = S + (size_t)b * LQ * LK;
  const int rowbase = it * 16 + 8 * (lane >> 4);
  const int col     = jt * 16 + (lane & 15);
#pragma unroll
  for (int r = 0; r < 8; r++) {
    Sb[(size_t)(rowbase + r) * LK + col] = acc[r] * SCALE_;
  }
}

// ---------------------------------------------------------------------------
// Phase 2: one wave32 per chain. With HAS_KEYS, iteration t's gather addresses
// are independent of iteration t-1 (keys from LDS, counter is loop index), so
// the unrolled loop keeps several steps' score/v gathers in flight.
// Mask: 4096 bits in 4 VGPRs/lane (word w -> lane w&31, slot w>>5), broadcast
// via __shfl. Lane l owns v-row components {2l, 2l+1}.
// ---------------------------------------------------------------------------
template <bool USE_S, bool HAS_KEYS>
__device__ __forceinline__ void run_chain(int c, int lane,
                                          const float* __restrict__ q,
                                          const float* __restrict__ k,
                                          const float* __restrict__ v,
                                          const float* __restrict__ S,
                                          const unsigned* __restrict__ skeys,
                                          float& out0, float& out1) {
  const int qrow = c >> 2;                     // c / RUNS
  const int b    = qrow >> 10;                 // / LQ
  const unsigned pair = (unsigned)(c & 8191);  // jax random_bits halves pairing
  const int hi = (c >> 13) & 1;                // which threefry output word

  const float* vbase = v + (size_t)b * LK * DD;
  const float* Srow  = USE_S ? (S + (size_t)qrow * LK) : nullptr;
  const float* kbase = k + (size_t)b * LK * DD;

  float q0 = 0.f, q1 = 0.f;
  if (!USE_S) {
    const float* qr = q + (size_t)qrow * DD;
    q0 = qr[2 * lane];
    q1 = qr[2 * lane + 1];
  }

  unsigned m0 = 0u, m1 = 0u, m2 = 0u, m3 = 0u;  // membership bitmask
  float s0 = 0.f, s1 = 0.f;
  int count = 0;
  unsigned rng0 = 0u, rng1 = 42u;               // used only when !HAS_KEYS

#pragma unroll 4
  for (int t = 0; t < STEPS_; t++) {
    unsigned ka0, ka1, kb0, kb1;
    if constexpr (HAS_KEYS) {
      ka0 = skeys[4 * t + 0];
      ka1 = skeys[4 * t + 1];
      kb0 = skeys[4 * t + 2];
      kb1 = skeys[4 * t + 3];
    } else {
      unsigned a0, a1, b0, b1, c0, c1;
      threefry2x32(rng0, rng1, 0u, 3u, a0, a1);
      threefry2x32(rng0, rng1, 1u, 4u, b0, b1);
      threefry2x32(rng0, rng1, 2u, 5u, c0, c1);
      ka0 = c0; ka1 = a1;
      kb0 = b1; kb1 = c1;
      rng0 = a0; rng1 = b0;
    }

    unsigned r0, r1;
    threefry2x32(ka0, ka1, pair, pair + 8192u, r0, r1);
    const unsigned vidx = (hi ? r1 : r0) & (unsigned)(LK - 1);  // randint % 4096

    // Issue gathers as early as possible (addresses independent of mask state).
    const float* vrow = vbase + (size_t)vidx * DD;
    const float v0 = vrow[2 * lane];
    const float v1 = vrow[2 * lane + 1];

    float a_v;
    if constexpr (USE_S) {
      a_v = Srow[vidx];                  // prescaled score (4B, L2-resident)
    } else {
      const float* krow = kbase + (size_t)vidx * DD;
      float part = q0 * krow[2 * lane] + q1 * krow[2 * lane + 1];
#pragma unroll
      for (int off = 16; off > 0; off >>= 1) part += __shfl_xor(part, off, 32);
      a_v = part * SCALE_;
    }

    threefry2x32(kb0, kb1, pair, pair + 8192u, r0, r1);
    const unsigned ub = hi ? r1 : r0;
    const float z = __uint_as_float((ub >> 9) | 0x3f800000u) - 1.0f;  // U[0,1)

    const float p = 1.0f / (1.0f + __expf(-a_v));
    const bool newin = (z <= p);

    const unsigned w    = vidx >> 5;
    const unsigned bit  = vidx & 31u;
    const unsigned slot = w >> 5;
    const int owner     = (int)(w & 31u);
    unsigned word = (slot == 0u) ? m0 : (slot == 1u) ? m1 : (slot == 2u) ? m2 : m3;
    word = (unsigned)__shfl((int)word, owner, 32);
    const unsigned oldin = (word >> bit) & 1u;
    const int sign = (int)newin - (int)oldin;
    const unsigned neww = newin ? (word | (1u << bit)) : (word & ~(1u << bit));
    if (lane == owner) {
      if (slot == 0u) m0 = neww;
      else if (slot == 1u) m1 = neww;
      else if (slot == 2u) m2 = neww;
      else m3 = neww;
    }

    count += sign;
    const float fs = (float)sign;       // -1, 0, +1
    s0 = fmaf(fs, v0, s0);
    s1 = fmaf(fs, v1, s1);
  }

  const float inv = 1.0f / fmaxf((float)count, 1e-8f);
  out0 = s0 * inv;
  out1 = s1 * inv;
}

template <bool USE_S, bool HAS_KEYS>
__global__ void gibbs_chain_kernel(const float* __restrict__ q,
                                   const float* __restrict__ k,
                                   const float* __restrict__ v,
                                   const float* __restrict__ S,
                                   const unsigned* __restrict__ keysG,
                                   float* __restrict__ chainOut) {
  __shared__ unsigned skeys[STEPS_ * 4];   // 1KB: per-step (ka0,ka1,kb0,kb1)
  if constexpr (HAS_KEYS) {
    if (threadIdx.x < STEPS_ * 4) skeys[threadIdx.x] = keysG[threadIdx.x];
    __syncthreads();
  }
  const int lane = (int)(threadIdx.x & 31u);
  const int c = blockIdx.x * (blockDim.x >> 5) + (int)(threadIdx.x >> 5);
  float o0, o1;
  run_chain<USE_S, HAS_KEYS>(c, lane, q, k, v, S, skeys, o0, o1);
  chainOut[(size_t)c * DD + 2 * lane]     = o0;
  chainOut[(size_t)c * DD + 2 * lane + 1] = o1;
}

__global__ void reduce_runs_kernel(const float* __restrict__ chainOut,
                                   float* __restrict__ out) {
  const int i = blockIdx.x * blockDim.x + (int)threadIdx.x;  // NROWS*DD total
  const int comp = i & (DD - 1);
  const int row  = i >> 6;
  const float* base = chainOut + (size_t)row * RUNS_ * DD + comp;
  out[i] = 0.25f * (base[0] + base[DD] + base[2 * DD] + base[3 * DD]);
}

// Fallback when workspace is too small for even the per-chain buffer:
// one wave per query row, loops over the 4 runs, writes output directly.
__global__ void gibbs_direct_kernel(const float* __restrict__ q,
                                    const float* __restrict__ k,
                                    const float* __restrict__ v,
                                    float* __restrict__ out) {
  const int lane = (int)(threadIdx.x & 31u);
  const int qrow = blockIdx.x * (blockDim.x >> 5) + (int)(threadIdx.x >> 5);
  float acc0 = 0.f, acc1 = 0.f;
  for (int r = 0; r < RUNS_; r++) {
    float o0, o1;
    run_chain<false, false>(qrow * RUNS_ + r, lane, q, k, v, nullptr, nullptr,
                            o0, o1);
    acc0 += o0;
    acc1 += o1;
  }
  out[(size_t)qrow * DD + 2 * lane]     = 0.25f * acc0;
  out[(size_t)qrow * DD + 2 * lane + 1] = 0.25f * acc1;
}

extern "C" void kernel_launch(void* const* d_in, const int* in_sizes, int n_in,
                              void* d_out, int out_size, void* d_ws, size_t ws_size,
                              hipStream_t stream) {
  (void)in_sizes; (void)n_in; (void)out_size;
  const float* q = (const float*)d_in[0];
  const float* k = (const float*)d_in[1];
  const float* v = (const float*)d_in[2];
  float* out = (float*)d_out;

  const size_t KEYS_BYTES  = (size_t)STEPS_ * 4 * sizeof(unsigned);     // 1 KB
  const size_t CHAIN_BYTES = (size_t)NCHAINS * DD * sizeof(float);      // 4 MB
  const size_t S_BYTES     = (size_t)B_ * LQ * LK * sizeof(float);      // 64 MB

  const int chainBlocks  = NCHAINS / 8;            // 8 waves (1 chain each)/block
  const int reduceBlocks = (NROWS * DD) / 256;     // 1024 blocks

  // ws layout: [keys 1KB][chain 4MB][S 64MB]
  unsigned* keys  = (unsigned*)d_ws;
  float*    chain = (float*)((char*)d_ws + KEYS_BYTES);
  float*    S     = (float*)((char*)d_ws + KEYS_BYTES + CHAIN_BYTES);

  if (ws_size >= KEYS_BYTES + CHAIN_BYTES + S_BYTES) {
    const int tiles = B_ * (LQ / 16) * (LK / 16);  // 65536 waves
    compute_keys_kernel<<<1, 32, 0, stream>>>(keys);
    scores_wmma_kernel<<<tiles / 8, 256, 0, stream>>>(q, k, S);
    gibbs_chain_kernel<true, true><<<chainBlocks, 256, 0, stream>>>(
        q, k, v, S, keys, chain);
    reduce_runs_kernel<<<reduceBlocks, 256, 0, stream>>>(chain, out);
  } else if (ws_size >= KEYS_BYTES + CHAIN_BYTES) {
    compute_keys_kernel<<<1, 32, 0, stream>>>(keys);
    gibbs_chain_kernel<false, true><<<chainBlocks, 256, 0, stream>>>(
        q, k, v, nullptr, keys, chain);
    reduce_runs_kernel<<<reduceBlocks, 256, 0, stream>>>(chain, out);
  } else {
    gibbs_direct_kernel<<<NROWS / 8, 256, 0, stream>>>(q, k, v, out);
  }
}